// GEN_14087492730940
// MI455X (gfx1250) — compile-verified
//
#include <hip/hip_runtime.h>
#include <hip/hip_bf16.h>

typedef float v2f __attribute__((ext_vector_type(2)));
typedef float v8f __attribute__((ext_vector_type(8)));

#define DIMD 128                    // feature dim (in/out)
#define DIMK 256                    // 2*D concat input
#define LDS_BYTES (128 * 128 * 8)   // 128 k-pairs x 128 n x float2 = 131072 B

// ---------------------------------------------------------------------------
// Zero the inbox scratch (d_ws is poisoned with 0xAA by the harness).
// ---------------------------------------------------------------------------
__global__ void zero_f32_kernel(float* __restrict__ p, long long n) {
    long long i = (long long)blockIdx.x * blockDim.x + threadIdx.x;
    long long stride = (long long)gridDim.x * blockDim.x;
    for (; i < n; i += stride) p[i] = 0.0f;
}

// ---------------------------------------------------------------------------
// Stage W^T into LDS as k-pair float2 entries: ldsW[(k>>1)*128 + n] =
// { W[n][2p], W[n][2p+1] } so a B fragment is one 8-byte LDS load.
// ---------------------------------------------------------------------------
__device__ __forceinline__ void stage_weights(const float* __restrict__ W, v2f* ldsW) {
    for (int i = threadIdx.x; i < 128 * 128; i += blockDim.x) {
        int p = i >> 7;       // k-pair index 0..127
        int n = i & 127;      // output feature
        v2f w = *(const v2f*)(W + (size_t)n * DIMK + 2 * p);
        ldsW[i] = w;          // contiguous LDS write, conflict-free
    }
    __syncthreads();
}

__device__ __forceinline__ void tile_coords(int tile, int tilesPerBatch, int& b, int& tb) {
    b = tile / tilesPerBatch;
    tb = (tile - b * tilesPerBatch) << 4;
}

// ---------------------------------------------------------------------------
// Compute TWO 16-row x 128-col tiles sharing every B fragment (weights are
// tile-invariant): halves LDS traffic per WMMA vs single-tile blocking.
// A fragment (f32 16x4): VGPR0 = K+0, VGPR1 = K+1; lane half selects K vs K+2.
// B fragment mirrors: VGPR j holds K = 4*kk + 2*half + j, N = lane&15.
// ---------------------------------------------------------------------------
__device__ __forceinline__ void gemm_tile_pair(const float* __restrict__ pR0,
                                               const float* __restrict__ pS0,
                                               const float* __restrict__ pR1,
                                               const float* __restrict__ pS1,
                                               const v2f* __restrict__ ldsW,
                                               const float biasv[8],
                                               int col, int half,
                                               v8f acc0[8], v8f acc1[8]) {
#pragma unroll
    for (int t = 0; t < 8; ++t) {
        v8f z;
#pragma unroll
        for (int j = 0; j < 8; ++j) z[j] = biasv[t];
        acc0[t] = z;
        acc1[t] = z;
    }
#pragma unroll 2
    for (int kk = 0; kk < 32; ++kk) {                 // receiver half of concat
        int kb = kk * 4 + 2 * half;
        v2f a0 = *(const v2f*)(pR0 + kb);
        v2f a1 = *(const v2f*)(pR1 + kb);
        const v2f* wrow = ldsW + (size_t)(kk * 2 + half) * 128;
#pragma unroll
        for (int t = 0; t < 8; ++t) {
            v2f bf = wrow[col + 16 * t];
            acc0[t] = __builtin_amdgcn_wmma_f32_16x16x4_f32(
                false, a0, false, bf, (short)0, acc0[t], false, false);
            acc1[t] = __builtin_amdgcn_wmma_f32_16x16x4_f32(
                false, a1, false, bf, (short)0, acc1[t], false, false);
        }
    }
#pragma unroll 2
    for (int kk = 0; kk < 32; ++kk) {                 // sender half of concat
        int kb = kk * 4 + 2 * half;
        v2f a0 = *(const v2f*)(pS0 + kb);
        v2f a1 = *(const v2f*)(pS1 + kb);
        const v2f* wrow = ldsW + (size_t)((kk + 32) * 2 + half) * 128;
#pragma unroll
        for (int t = 0; t < 8; ++t) {
            v2f bf = wrow[col + 16 * t];
            acc0[t] = __builtin_amdgcn_wmma_f32_16x16x4_f32(
                false, a0, false, bf, (short)0, acc0[t], false, false);
            acc1[t] = __builtin_amdgcn_wmma_f32_16x16x4_f32(
                false, a1, false, bf, (short)0, acc1[t], false, false);
        }
    }
}

// ---------------------------------------------------------------------------
// In-register LayerNorm stats. C/D layout: VGPR v, lane half h -> row M=v+8h,
// col N = (lane&15)+16t.  Butterfly over xor masks 1,2,4,8 reduces the 16
// lanes of each half (full 128-col row = 8 tiles x 16 lanes).
// ---------------------------------------------------------------------------
__device__ __forceinline__ void ln_stats(const v8f acc[8], float mu[8], float rsig[8]) {
#pragma unroll
    for (int v = 0; v < 8; ++v) {
        float s = 0.0f, q = 0.0f;
#pragma unroll
        for (int t = 0; t < 8; ++t) { float x = acc[t][v]; s += x; q += x * x; }
#pragma unroll
        for (int m = 1; m <= 8; m <<= 1) {
            s += __shfl_xor(s, m, 32);
            q += __shfl_xor(q, m, 32);
        }
        float mean = s * (1.0f / 128.0f);
        float var  = fmaxf(q * (1.0f / 128.0f) - mean * mean, 0.0f);
        mu[v]   = mean;
        rsig[v] = rsqrtf(var + 1e-5f);
    }
}

// ---------------------------------------------------------------------------
// Edge epilogue: LayerNorm + atomic scatter-add into inbox[b, recv, :].
// For fixed (v,t) the 16 lanes of a half hit 16 consecutive floats of one
// inbox row -> cacheline-contiguous atomics, L2-resident target.
// ---------------------------------------------------------------------------
__device__ __forceinline__ void edge_writeback(const v8f acc[8],
                                               const int* __restrict__ receivers,
                                               float* __restrict__ inbox,
                                               const float gv[8], const float bv[8],
                                               int b, int tb, int Nn, int col, int half) {
    float mu[8], rsig[8];
    ln_stats(acc, mu, rsig);
#pragma unroll
    for (int v = 0; v < 8; ++v) {
        int e = tb + v + 8 * half;                    // row M -> edge id
        int r = receivers[e];
        float* dst = inbox + ((size_t)b * Nn + r) * DIMD + col;
#pragma unroll
        for (int t = 0; t < 8; ++t) {
            float val = (acc[t][v] - mu[v]) * rsig[v] * gv[t] + bv[t];
            atomicAdd(dst + 16 * t, val);
        }
    }
}

// ---------------------------------------------------------------------------
// Kernel 1: per-edge messages = LN(Wm @ [x_recv; x_send] + bm), scatter-added
// into the inbox with global float atomics.  Two edge tiles per wave.
// ---------------------------------------------------------------------------
__global__ void __launch_bounds__(256)
edge_msg_kernel(const float* __restrict__ nodes, const float* __restrict__ Wm,
                const float* __restrict__ bm, const float* __restrict__ g1,
                const float* __restrict__ b1, const int* __restrict__ senders,
                const int* __restrict__ receivers, float* __restrict__ inbox,
                int Bn, int Nn, int En) {
    extern __shared__ v2f ldsW[];
    stage_weights(Wm, ldsW);

    const int lane = threadIdx.x & 31;
    const int half = lane >> 4;
    const int col  = lane & 15;
    const int wavesPerBlock = blockDim.x >> 5;
    const int gwave  = blockIdx.x * wavesPerBlock + (threadIdx.x >> 5);
    const int nwaves = gridDim.x * wavesPerBlock;

    float gv[8], bv[8], biasv[8];
#pragma unroll
    for (int t = 0; t < 8; ++t) {
        int n = col + 16 * t;
        gv[t] = g1[n]; bv[t] = b1[n]; biasv[t] = bm[n];
    }

    const int tilesPerBatch = En >> 4;
    const int totalPairs = (Bn * tilesPerBatch) >> 1;   // total tiles even

    for (int pair = gwave; pair < totalPairs; pair += nwaves) {
        int b0, tb0, b1i, tb1;
        tile_coords(2 * pair,     tilesPerBatch, b0,  tb0);
        tile_coords(2 * pair + 1, tilesPerBatch, b1i, tb1);

        const float* pR0 = nodes + ((size_t)b0  * Nn + receivers[tb0 + col]) * DIMD;
        const float* pS0 = nodes + ((size_t)b0  * Nn + senders[tb0 + col])   * DIMD;
        const float* pR1 = nodes + ((size_t)b1i * Nn + receivers[tb1 + col]) * DIMD;
        const float* pS1 = nodes + ((size_t)b1i * Nn + senders[tb1 + col])   * DIMD;

        v8f acc0[8], acc1[8];
        gemm_tile_pair(pR0, pS0, pR1, pS1, ldsW, biasv, col, half, acc0, acc1);

        edge_writeback(acc0, receivers, inbox, gv, bv, b0,  tb0, Nn, col, half);
        edge_writeback(acc1, receivers, inbox, gv, bv, b1i, tb1, Nn, col, half);
    }
}

// ---------------------------------------------------------------------------
// Node epilogue: LayerNorm + direct store (lane groups write 64B contiguous).
// ---------------------------------------------------------------------------
__device__ __forceinline__ void node_writeback(const v8f acc[8], float* __restrict__ out,
                                               const float gv[8], const float bv[8],
                                               int b, int tb, int Nn, int col, int half) {
    float mu[8], rsig[8];
    ln_stats(acc, mu, rsig);
#pragma unroll
    for (int v = 0; v < 8; ++v) {
        int nrow = tb + v + 8 * half;
        float* dst = out + ((size_t)b * Nn + nrow) * DIMD + col;
#pragma unroll
        for (int t = 0; t < 8; ++t)
            dst[16 * t] = (acc[t][v] - mu[v]) * rsig[v] * gv[t] + bv[t];
    }
}

// ---------------------------------------------------------------------------
// Kernel 2: out = LN(Wn @ [nodes; inbox] + bn).  Two node tiles per wave.
// ---------------------------------------------------------------------------
__global__ void __launch_bounds__(256)
node_out_kernel(const float* __restrict__ nodes, const float* __restrict__ inbox,
                const float* __restrict__ Wn, const float* __restrict__ bn,
                const float* __restrict__ g2, const float* __restrict__ b2,
                float* __restrict__ out, int Bn, int Nn) {
    extern __shared__ v2f ldsW[];
    stage_weights(Wn, ldsW);

    const int lane = threadIdx.x & 31;
    const int half = lane >> 4;
    const int col  = lane & 15;
    const int wavesPerBlock = blockDim.x >> 5;
    const int gwave  = blockIdx.x * wavesPerBlock + (threadIdx.x >> 5);
    const int nwaves = gridDim.x * wavesPerBlock;

    float gv[8], bv[8], biasv[8];
#pragma unroll
    for (int t = 0; t < 8; ++t) {
        int n = col + 16 * t;
        gv[t] = g2[n]; bv[t] = b2[n]; biasv[t] = bn[n];
    }

    const int tilesPerBatch = Nn >> 4;
    const int totalPairs = (Bn * tilesPerBatch) >> 1;   // 5000 tiles -> 2500 pairs

    for (int pair = gwave; pair < totalPairs; pair += nwaves) {
        int b0, tb0, b1i, tb1;
        tile_coords(2 * pair,     tilesPerBatch, b0,  tb0);
        tile_coords(2 * pair + 1, tilesPerBatch, b1i, tb1);

        size_t row0 = (size_t)b0  * Nn + (tb0 + col);
        size_t row1 = (size_t)b1i * Nn + (tb1 + col);
        const float* pR0 = nodes + row0 * DIMD;
        const float* pS0 = inbox + row0 * DIMD;
        const float* pR1 = nodes + row1 * DIMD;
        const float* pS1 = inbox + row1 * DIMD;

        v8f acc0[8], acc1[8];
        gemm_tile_pair(pR0, pS0, pR1, pS1, ldsW, biasv, col, half, acc0, acc1);

        node_writeback(acc0, out, gv, bv, b0,  tb0, Nn, col, half);
        node_writeback(acc1, out, gv, bv, b1i, tb1, Nn, col, half);
    }
}

// ---------------------------------------------------------------------------
extern "C" void kernel_launch(void* const* d_in, const int* in_sizes, int n_in,
                              void* d_out, int out_size, void* d_ws, size_t ws_size,
                              hipStream_t stream) {
    const float* nodes = (const float*)d_in[0];
    const float* Wm    = (const float*)d_in[1];
    const float* bm    = (const float*)d_in[2];
    const float* g1    = (const float*)d_in[3];
    const float* b1    = (const float*)d_in[4];
    const float* Wn    = (const float*)d_in[5];
    const float* bn    = (const float*)d_in[6];
    const float* g2    = (const float*)d_in[7];
    const float* b2    = (const float*)d_in[8];
    const int* senders   = (const int*)d_in[9];
    const int* receivers = (const int*)d_in[10];
    float* out = (float*)d_out;

    const int B = 8;
    const int E = in_sizes[9];                 // 160000
    const int N = (in_sizes[0] / DIMD) / B;    // 10000

    float* inbox = (float*)d_ws;               // B*N*128 f32 = 40.96 MB
    long long inboxElems = (long long)B * N * DIMD;

    (void)hipFuncSetAttribute((const void*)edge_msg_kernel,
                              hipFuncAttributeMaxDynamicSharedMemorySize, LDS_BYTES);
    (void)hipFuncSetAttribute((const void*)node_out_kernel,
                              hipFuncAttributeMaxDynamicSharedMemorySize, LDS_BYTES);

    zero_f32_kernel<<<2048, 256, 0, stream>>>(inbox, inboxElems);

    edge_msg_kernel<<<1024, 256, LDS_BYTES, stream>>>(
        nodes, Wm, bm, g1, b1, senders, receivers, inbox, B, N, E);

    node_out_kernel<<<640, 256, LDS_BYTES, stream>>>(
        nodes, inbox, Wn, bn, g2, b2, out, B, N);
}